// OneLinkRopeDynamics_60679297957949
// MI455X (gfx1250) — compile-verified
//
#include <hip/hip_runtime.h>
#include <cstdint>

#define QEPS 0.001f
#define ROWS_PER_BLOCK 256

// ---------------------------------------------------------------------------
// CDNA5 async global<->LDS DMA (ASYNCcnt path, gfx1250-specific)
// ---------------------------------------------------------------------------
__device__ __forceinline__ uint32_t lds_off_of(const void* p) {
  // Generic pointers into the LDS aperture carry the LDS byte offset in the
  // low 32 bits (aperture tag lives in [63:32]).
  return (uint32_t)(uintptr_t)p;
}

__device__ __forceinline__ void async_load_b128(uint32_t lds_byte_addr,
                                                const void* saddr,
                                                uint32_t byte_off) {
  // GVS mode: addr = SGPR64 + VGPR32 offset. Streams 16B/lane straight to LDS.
  asm volatile("global_load_async_to_lds_b128 %0, %1, %2 th:TH_LOAD_NT"
               :: "v"(lds_byte_addr), "v"(byte_off), "s"(saddr)
               : "memory");
}

__device__ __forceinline__ void async_store_b128(uint32_t lds_byte_addr,
                                                 void* saddr,
                                                 uint32_t byte_off) {
  asm volatile("global_store_async_from_lds_b128 %1, %0, %2 th:TH_STORE_NT"
               :: "v"(lds_byte_addr), "v"(byte_off), "s"(saddr)
               : "memory");
}

__device__ __forceinline__ void wait_async0() {
  asm volatile("s_wait_asynccnt 0" ::: "memory");
}

// ---------------------------------------------------------------------------
// Single-instruction hardware transcendentals (v_sqrt_f32 / v_rsq_f32 / v_rcp)
// ---------------------------------------------------------------------------
__device__ __forceinline__ float hsqrt(float x)  { return __builtin_amdgcn_sqrtf(x); }
__device__ __forceinline__ float hrsqrt(float x) { return __builtin_amdgcn_rsqf(x); }
__device__ __forceinline__ float hrcp(float x)   { return __builtin_amdgcn_rcpf(x); }

// ---------------------------------------------------------------------------
// Per-row math. All inverse-trig + trig pairs are folded with half-angle
// identities (every angle is only ever consumed as sin/cos of angle/2), so
// the whole row costs ~120 VALU + a handful of v_sqrt/v_rsq — memory-bound.
// ---------------------------------------------------------------------------
__device__ __forceinline__ void qrot(float w, float x, float y, float z,
                                     float vx, float vy, float vz,
                                     float& rx, float& ry, float& rz) {
  float ux  = y * vz - z * vy;
  float uy  = z * vx - x * vz;
  float uz  = x * vy - y * vx;
  float uux = y * uz - z * uy;
  float uuy = z * ux - x * uz;
  float uuz = x * uy - y * ux;
  rx = vx + 2.0f * (w * ux + uux);
  ry = vy + 2.0f * (w * uy + uuy);
  rz = vz + 2.0f * (w * uz + uuz);
}

__device__ __forceinline__ void compute_row(const float* st, const float* ac,
                                            float* o) {
  const float px = st[0], py = st[1], pz = st[2];
  const float qw = st[3], qx = st[4], qy = st[5], qz = st[6];

  // qinverse
  const float iw = qw, ix = -qx, iy = -qy, iz = -qz;

  // grips in body frame
  float g1x, g1y, g1z, g2x, g2y, g2z;
  qrot(iw, ix, iy, iz, ac[0], ac[1], ac[2], g1x, g1y, g1z);
  qrot(iw, ix, iy, iz, ac[3], ac[4], ac[5], g2x, g2y, g2z);
  g1x -= 0.5f;
  g2x += 0.5f;

  const float prx = 0.5f * (g1x + g2x);
  const float pry = 0.5f * (g1y + g2y);
  const float prz = 0.5f * (g1z + g2z);

  const float dx = g2x - prx, dy = g2y - pry, dz = g2z - prz;

  // ---- rotation_between_two_vectors((1,0,0), d) ----
  const float dinv = hrsqrt(dx * dx + dy * dy + dz * dz);
  const float v2x = dx * dinv, v2y = dy * dinv, v2z = dz * dinv;
  const float dot = fminf(fmaxf(v2x, -1.0f + QEPS), 1.0f - QEPS);

  // axis_pre = cross((1,0,0), v2) = (0, -v2z, v2y)  -> axis.x == 0 always
  const float apy = -v2z, apz = v2y;
  const float an  = hsqrt(apy * apy + apz * apz);
  const bool small = an < QEPS;
  const float ainv = small ? 0.0f : hrcp(an);
  const float axy = apy * ainv, axz = apz * ainv;

  // angle = acos(dot) in [0,pi]; half-angle without trig:
  //   cos(angle/2) = sqrt((1+dot)/2), sin(angle/2) = sqrt((1-dot)/2)
  float r0 = small ? 1.0f : hsqrt((1.0f + dot) * 0.5f);
  float sh = small ? 0.0f : hsqrt((1.0f - dot) * 0.5f);
  float r2 = axy * sh, r3 = axz * sh;           // r1 == 0 identically
  const float rn = hrsqrt(r0 * r0 + r2 * r2 + r3 * r3);
  r0 *= rn; r2 *= rn; r3 *= rn;                 // bdot(q,q)>0 post-normalize

  // ---- new_pos = pos + qrot(quat, new_pos_rel) ----
  float rpx, rpy, rpz;
  qrot(qw, qx, qy, qz, prx, pry, prz, rpx, rpy, rpz);

  // ---- new_quat_pre = qmul(quat, (r0, 0, r2, r3)) ----
  const float p0 = qw * r0 - qy * r2 - qz * r3;
  const float p1 = qx * r0 + qy * r3 - qz * r2;
  const float p2 = qw * r2 - qx * r3 + qy * r0;
  const float p3 = qw * r3 + qx * r2 + qz * r0;

  // ---- qeuler_zyx * (0,1,1), then eulerq_zyx, trig-free ----
  // ye = asin(sarg), ye/2 in [-pi/4, pi/4]:
  //   cos(ye) = sqrt(1-sarg^2); cy = sqrt((1+cos ye)/2) >= 0
  //   sy = sign(sarg) * sqrt((1-cos ye)/2)
  const float sarg = fminf(fmaxf(2.0f * (p1 * p3 + p0 * p2), -1.0f), 1.0f);
  const float cye  = hsqrt(fmaxf(1.0f - sarg * sarg, 0.0f));
  const float cy   = hsqrt((1.0f + cye) * 0.5f);
  const float sy   = copysignf(hsqrt(fmaxf((1.0f - cye) * 0.5f, 0.0f)), sarg);

  // ze = atan2(za, zb), ze/2 in (-pi/2, pi/2] so cz >= 0;
  //   sz sign follows sign(za) (signed zero reproduces the ze = +/-pi branch)
  const float za  = 2.0f * (p0 * p3 - p1 * p2);
  const float zb  = 1.0f - 2.0f * (p2 * p2 + p3 * p3);
  const float zh  = hrsqrt(za * za + zb * zb);
  const float cze = zb * zh;
  const float cz  = hsqrt(fmaxf((1.0f + cze) * 0.5f, 0.0f));
  const float sz  = copysignf(hsqrt(fmaxf((1.0f - cze) * 0.5f, 0.0f)), za);

  // new_quat = qmul(rz, ry)  (rx is identity since x-euler is zeroed)
  o[0] = px + rpx;
  o[1] = py + rpy;
  o[2] = pz + rpz;
  o[3] = cz * cy;
  o[4] = -sz * sy;
  o[5] = cz * sy;
  o[6] = sz * cy;
}

// ---------------------------------------------------------------------------
// Kernel: block-tile async DMA staging through LDS, per-row compute.
// ---------------------------------------------------------------------------
__global__ __launch_bounds__(ROWS_PER_BLOCK) void
OneLinkRopeDynamics_kernel(const float* __restrict__ state,
                           const float* __restrict__ action,
                           float* __restrict__ out, int N) {
  __shared__ __align__(16) float s_state[ROWS_PER_BLOCK * 7];   // 7168 B
  __shared__ __align__(16) float s_action[ROWS_PER_BLOCK * 6];  // 6144 B

  const int tid = threadIdx.x;
  const long long base = (long long)blockIdx.x * ROWS_PER_BLOCK;

  if (base + ROWS_PER_BLOCK <= (long long)N) {
    // ---- full tile: async DMA global -> LDS (coalesced b128 streams) ----
    const uint32_t sb = (uint32_t)(base * 28);  // byte base of state tile
    const uint32_t ab = (uint32_t)(base * 24);  // byte base of action tile
    const uint32_t lds_s = lds_off_of(s_state);
    const uint32_t lds_a = lds_off_of(s_action);

    // state tile: 7168 B = 448 x b128
    for (int idx = tid; idx < 448; idx += ROWS_PER_BLOCK)
      async_load_b128(lds_s + idx * 16, state, sb + idx * 16);
    // action tile: 6144 B = 384 x b128
    for (int idx = tid; idx < 384; idx += ROWS_PER_BLOCK)
      async_load_b128(lds_a + idx * 16, action, ab + idx * 16);

    wait_async0();       // this wave's DMA done (LDS writes visible)
    __syncthreads();     // every wave's DMA done

    float st[7], ac[6], o[7];
#pragma unroll
    for (int k = 0; k < 7; ++k) st[k] = s_state[tid * 7 + k];
#pragma unroll
    for (int k = 0; k < 6; ++k) ac[k] = s_action[tid * 6 + k];

    compute_row(st, ac, o);

    // results back into the (same-thread) state slab, then DMA out
#pragma unroll
    for (int k = 0; k < 7; ++k) s_state[tid * 7 + k] = o[k];
    __syncthreads();

    for (int idx = tid; idx < 448; idx += ROWS_PER_BLOCK)
      async_store_b128(lds_s + idx * 16, out, sb + idx * 16);
    wait_async0();
  } else {
    // ---- partial tail tile: plain per-row path ----
    const long long row = base + tid;
    if (row < (long long)N) {
      float st[7], ac[6], o[7];
#pragma unroll
      for (int k = 0; k < 7; ++k) st[k] = state[row * 7 + k];
#pragma unroll
      for (int k = 0; k < 6; ++k) ac[k] = action[row * 6 + k];
      compute_row(st, ac, o);
#pragma unroll
      for (int k = 0; k < 7; ++k) out[row * 7 + k] = o[k];
    }
  }
}

// ---------------------------------------------------------------------------
extern "C" void kernel_launch(void* const* d_in, const int* in_sizes, int n_in,
                              void* d_out, int out_size, void* d_ws,
                              size_t ws_size, hipStream_t stream) {
  (void)n_in; (void)out_size; (void)d_ws; (void)ws_size;
  const float* state  = (const float*)d_in[0];
  const float* action = (const float*)d_in[1];
  float* out = (float*)d_out;

  const int N = in_sizes[0] / 7;
  const int grid = (N + ROWS_PER_BLOCK - 1) / ROWS_PER_BLOCK;

  OneLinkRopeDynamics_kernel<<<grid, ROWS_PER_BLOCK, 0, stream>>>(state, action,
                                                                  out, N);
}